// _SupervisedGCN_4080218931849
// MI455X (gfx1250) — compile-verified
//
#include <hip/hip_runtime.h>

typedef __attribute__((ext_vector_type(2))) float v2f;
typedef __attribute__((ext_vector_type(8))) float v8f;

typedef int v4i_t __attribute__((vector_size(16)));
typedef __attribute__((address_space(3))) v4i_t* lds_v4i_p;

#define K_DIM 128

#if __has_builtin(__builtin_amdgcn_global_load_async_to_lds_b128)
#define HAVE_ASYNC_LDS 1
#else
#define HAVE_ASYNC_LDS 0
#endif

// ---------------- degree / normalization ----------------

__global__ void deg_init_kernel(float* deg, int N) {
  int i = blockIdx.x * blockDim.x + threadIdx.x;
  if (i < N) deg[i] = 1.0f;  // self-loop contribution
}

__global__ void deg_accum_kernel(const int* __restrict__ dst, float* deg, int E) {
  int e = blockIdx.x * blockDim.x + threadIdx.x;
  if (e < E) atomicAdd(&deg[dst[e]], 1.0f);
}

__global__ void deg_rsqrt_kernel(float* deg, int N) {
  int i = blockIdx.x * blockDim.x + threadIdx.x;
  if (i < N) deg[i] = rsqrtf(deg[i]);  // deg > 0 always (self-loops)
}

// ---------------- WMMA f32 GEMM: C[M,NO] = X[M,128] * W[128,NO] ----------------
// One wave per 16x16 C tile; NO/16 waves per block; block covers one 16-row stripe.
// X tile staged to LDS via CDNA5 async global->LDS DMA (no VGPR round-trip).

template <int NO, bool RELU_IN>
__global__ __launch_bounds__(NO * 2) void gemm_wmma_kernel(
    const float* __restrict__ X, const float* __restrict__ W,
    float* __restrict__ C, int M) {
  constexpr int NWAVES = NO / 16;
  __shared__ float Xs[16][K_DIM + 4];  // +4 pad: stride 132 -> conflict-free A reads
  const int tid  = threadIdx.x;
  const int wave = tid >> 5;
  const int lane = tid & 31;
  const int m0   = blockIdx.x << 4;

  const float* xt = X + (size_t)m0 * K_DIM;
  const bool full_tile = (m0 + 16 <= M);

#if HAVE_ASYNC_LDS
  if (full_tile) {
    // 16 rows x 32 float4 chunks; per-lane 16B async DMA straight into LDS
    for (int j = tid; j < 16 * (K_DIM / 4); j += NWAVES * 32) {
      int r = j >> 5, c4 = j & 31;
      __builtin_amdgcn_global_load_async_to_lds_b128(
          (v4i_t*)(xt + (size_t)j * 4), (lds_v4i_p)&Xs[r][c4 << 2], 0, 0);
    }
#if __has_builtin(__builtin_amdgcn_s_wait_asynccnt)
    __builtin_amdgcn_s_wait_asynccnt(0);
#else
    asm volatile("s_wait_asynccnt 0" ::: "memory");
#endif
  } else
#endif
  {
    if (full_tile) {
      // vectorized fallback staging
      for (int j = tid; j < 16 * (K_DIM / 4); j += NWAVES * 32) {
        int r = j >> 5, c4 = j & 31;
        float4 v = ((const float4*)xt)[j];
        *(float4*)&Xs[r][c4 << 2] = v;
      }
    } else {
      for (int i = tid; i < 16 * K_DIM; i += NWAVES * 32) {
        int r = i >> 7, c = i & (K_DIM - 1);
        Xs[r][c] = (m0 + r < M) ? xt[i] : 0.0f;
      }
    }
  }
  __syncthreads();

  const int n0    = wave << 4;
  const int l15   = lane & 15;          // A row / B,C col
  const int khalf = (lane >> 4) << 1;   // K sub-pair: lanes 0-15 -> K+0/1, 16-31 -> K+2/3
  v8f acc = {};                          // bias is added in the aggregation init pass

#pragma unroll
  for (int k = 0; k < K_DIM; k += 4) {
    v2f a, b;
    a.x = Xs[l15][k + khalf];
    a.y = Xs[l15][k + khalf + 1];
    if (RELU_IN) {  // previous layer's ReLU fused at fragment-read time
      a.x = fmaxf(a.x, 0.0f);
      a.y = fmaxf(a.y, 0.0f);
    }
    const float* wp = W + (size_t)(k + khalf) * NO + n0 + l15;
    b.x = wp[0];
    b.y = wp[NO];
    // (neg_a, A, neg_b, B, c_mod, C, reuse_a, reuse_b)
    acc = __builtin_amdgcn_wmma_f32_16x16x4_f32(false, a, false, b, (short)0, acc,
                                                false, false);
  }

  // C/D layout: VGPR v -> row v + 8*lane[4], col = lane&15
  float* cp = C + (size_t)m0 * NO + n0 + l15;
  const int rbase = (lane >> 4) << 3;
#pragma unroll
  for (int v = 0; v < 8; ++v) {
    if (m0 + rbase + v < M) cp[(size_t)(rbase + v) * NO] = acc[v];
  }
}

// ---------------- aggregation ----------------
// out[i] = bias + dis[i]^2 * h[i]   (self-loop + bias init, overwrites poison)

template <int NO>
__global__ void agg_init_kernel(const float* __restrict__ h, const float* __restrict__ dis,
                                const float* __restrict__ bias, float* __restrict__ out,
                                int N) {
  constexpr int L = NO / 4;
  int idx = blockIdx.x * blockDim.x + threadIdx.x;
  if (idx >= N * L) return;
  int i = idx / L, l = idx % L;
  float d = dis[i];
  float w = d * d;
  float4 v = ((const float4*)(h + (size_t)i * NO))[l];
  float4 b = ((const float4*)bias)[l];
  float4 r = make_float4(b.x + w * v.x, b.y + w * v.y, b.z + w * v.z, b.w + w * v.w);
  ((float4*)(out + (size_t)i * NO))[l] = r;
}

// out[dst] += dis[src]*dis[dst] * h[src]; NO/4 lanes per edge, float4 gathers

template <int NO>
__global__ void agg_edge_kernel(const float* __restrict__ h, const int* __restrict__ src,
                                const int* __restrict__ dst, const float* __restrict__ dis,
                                float* out, int E) {
  constexpr int L = NO / 4;
  long long idx = (long long)blockIdx.x * blockDim.x + threadIdx.x;
  int e = (int)(idx / L);
  int l = (int)(idx % L);
  if (e >= E) return;
  int s = src[e], d = dst[e];
  float w = dis[s] * dis[d];
  float4 v = ((const float4*)(h + (size_t)s * NO))[l];
  float* o = out + (size_t)d * NO + (l << 2);
  atomicAdd(o + 0, w * v.x);
  atomicAdd(o + 1, w * v.y);
  atomicAdd(o + 2, w * v.z);
  atomicAdd(o + 3, w * v.w);
}

// ---------------- driver ----------------

extern "C" void kernel_launch(void* const* d_in, const int* in_sizes, int n_in,
                              void* d_out, int out_size, void* d_ws, size_t ws_size,
                              hipStream_t stream) {
  const float* x  = (const float*)d_in[0];
  const int*   ei = (const int*)d_in[1];
  const float* W1 = (const float*)d_in[2];
  const float* b1 = (const float*)d_in[3];
  const float* W2 = (const float*)d_in[4];
  const float* b2 = (const float*)d_in[5];
  const float* W3 = (const float*)d_in[6];
  const float* b3 = (const float*)d_in[7];
  float* out = (float*)d_out;

  const int N = in_sizes[0] / K_DIM;
  const int E = in_sizes[1] / 2;
  const int* src = ei;
  const int* dst = ei + E;

  char* ws = (char*)d_ws;
  size_t off = 0;
  float* dis = (float*)(ws + off); off += ((size_t)N * 4 + 511) & ~(size_t)511;
  float* h1  = (float*)(ws + off); off += (size_t)N * K_DIM * 4;
  float* h2  = (float*)(ws + off);

  const int B = 256;
  deg_init_kernel<<<(N + B - 1) / B, B, 0, stream>>>(dis, N);
  deg_accum_kernel<<<(E + B - 1) / B, B, 0, stream>>>(dst, dis, E);
  deg_rsqrt_kernel<<<(N + B - 1) / B, B, 0, stream>>>(dis, N);

  const int mtiles = (N + 15) / 16;
  const int eg128 = (int)(((long long)E * 32 + B - 1) / B);
  const int eg64  = (int)(((long long)E * 16 + B - 1) / B);

  // layer 1: h1 = x@W1 ; h2 = agg(h1) + b1
  gemm_wmma_kernel<128, false><<<mtiles, 256, 0, stream>>>(x, W1, h1, N);
  agg_init_kernel<128><<<(N * 32 + B - 1) / B, B, 0, stream>>>(h1, dis, b1, h2, N);
  agg_edge_kernel<128><<<eg128, B, 0, stream>>>(h1, src, dst, dis, h2, E);

  // layer 2: h1 = relu(h2)@W2 ; h2 = agg(h1) + b2
  gemm_wmma_kernel<128, true><<<mtiles, 256, 0, stream>>>(h2, W2, h1, N);
  agg_init_kernel<128><<<(N * 32 + B - 1) / B, B, 0, stream>>>(h1, dis, b2, h2, N);
  agg_edge_kernel<128><<<eg128, B, 0, stream>>>(h1, src, dst, dis, h2, E);

  // layer 3: h1 = relu(h2)@W3 ; out = agg(h1) + b3
  gemm_wmma_kernel<64, true><<<mtiles, 128, 0, stream>>>(h2, W3, h1, N);
  agg_init_kernel<64><<<(N * 16 + B - 1) / B, B, 0, stream>>>(h1, dis, b3, out, N);
  agg_edge_kernel<64><<<eg64, B, 0, stream>>>(h1, src, dst, dis, out, E);
}